// Attention_77833397338144
// MI455X (gfx1250) — compile-verified
//
#include <hip/hip_runtime.h>
#include <hip/hip_bf16.h>
#include <math.h>

typedef __attribute__((ext_vector_type(2))) float v2f;
typedef __attribute__((ext_vector_type(8))) float v8f;

// ---------------------------------------------------------------------------
// GEMM: C[m][n] = sum_k A[m*K + k] * W[n*K + k]
// (A row-major [M,K], W row-major [N,K] i.e. torch nn.Linear weight layout)
// One wave32 computes one 16x16 fp32 C tile with V_WMMA_F32_16X16X4_F32.
// fp32 WMMA lane layout (ISA 7.12.2):
//   A 16x4 : lanes<16 -> K = k0+{0,1}, lanes>=16 -> K = k0+{2,3} (contiguous pair)
//   B 4x16 : symmetric (N = lane&15, same K split)  -> both are float2 loads
//   C 16x16: VGPR i: lanes0-15 -> (M=i, N=lane), lanes16-31 -> (M=8+i, N=lane-16)
// ---------------------------------------------------------------------------
__global__ __launch_bounds__(256) void gemm_awt_f32_wmma(
    const float* __restrict__ A, const float* __restrict__ W,
    float* __restrict__ C, int M, int N, int K)
{
    const int lane   = threadIdx.x & 31;
    const int waveId = threadIdx.x >> 5;
    const int tilesN = N >> 4;
    const int tile   = blockIdx.x * 8 + waveId;      // 8 waves per block
    const int tm     = tile / tilesN;
    const int tn     = tile - tm * tilesN;
    const int mn     = lane & 15;
    const int khalf  = (lane >> 4) << 1;             // 0 or 2

    const float* arow = A + (size_t)(tm * 16 + mn) * K + khalf;
    const float* wrow = W + (size_t)(tn * 16 + mn) * K + khalf;

    v8f acc = {};
    for (int k0 = 0; k0 < K; k0 += 4) {
        v2f a = *(const v2f*)(arow + k0);
        v2f b = *(const v2f*)(wrow + k0);
        acc = __builtin_amdgcn_wmma_f32_16x16x4_f32(
            /*neg_a=*/false, a, /*neg_b=*/false, b,
            /*c_mod=*/(short)0, acc, /*reuse_a=*/false, /*reuse_b=*/false);
    }

    const int row0 = tm * 16 + ((lane >> 4) << 3);   // +8 for upper half-wave
    const int col  = tn * 16 + mn;
    #pragma unroll
    for (int i = 0; i < 8; ++i)
        C[(size_t)(row0 + i) * N + col] = acc[i];
}

// ---------------------------------------------------------------------------
// scores[b,t,s] = sum_d v[d] * tanh(e1[b,s,d] + e2[b,t,d]); mask; softmax over S
// One block per (b,t). e2 row and v live in LDS. Lanes split D in float4
// chunks: one global_load_b128 per wave covers 512 contiguous bytes of e1,
// LDS traffic is ds_load_b128. tanh via v_exp_f32 + v_rcp_f32; wave shfl
// reduction; block-wide masked softmax.
// ---------------------------------------------------------------------------
__global__ __launch_bounds__(256) void attn_tanh_softmax(
    const float* __restrict__ e1,    // [B*S, D]
    const float* __restrict__ e2,    // [B*T, D]
    const int*   __restrict__ mask,  // [B, S]
    const float* __restrict__ v,     // [D]
    float* __restrict__ out)         // [B, T, S]
{
    constexpr int T = 256, S = 512, D = 512;
    __shared__ float e2row[D];
    __shared__ float vv[D];
    __shared__ float sc[S];
    __shared__ float red[256];

    const int tid  = threadIdx.x;
    const int lane = tid & 31;
    const int wid  = tid >> 5;
    const int bt   = blockIdx.x;
    const int b    = bt / T;

    // stage e2 row + v into LDS (float4 copies)
    {
        const float4* e2p = (const float4*)(e2 + (size_t)bt * D);
        const float4* vp  = (const float4*)v;
        float4* e2s = (float4*)e2row;
        float4* vs  = (float4*)vv;
        for (int i = tid; i < D / 4; i += 256) { e2s[i] = e2p[i]; vs[i] = vp[i]; }
    }
    __syncthreads();

    const float TWO_LOG2E = 2.8853900817779268f;   // 2*log2(e)
    const float4* e2s = (const float4*)e2row;
    const float4* vs  = (const float4*)vv;

    for (int s = wid; s < S; s += 8) {
        const float4* e1p = (const float4*)(e1 + (size_t)(b * S + s) * D);
        float partial = 0.f;
        #pragma unroll
        for (int j = 0; j < 4; ++j) {
            const int idx = lane + (j << 5);       // float4 index: 0..127
            float4 x1 = e1p[idx];                  // global_load_b128, coalesced
            float4 x2 = e2s[idx];                  // ds_load_b128
            float4 vx = vs[idx];                   // ds_load_b128
            float xs[4] = {x1.x + x2.x, x1.y + x2.y, x1.z + x2.z, x1.w + x2.w};
            float vvx[4] = {vx.x, vx.y, vx.z, vx.w};
            #pragma unroll
            for (int c = 0; c < 4; ++c) {
                float x = fminf(fmaxf(xs[c], -10.f), 10.f);  // keep e^{2x} finite
                float ex = __builtin_amdgcn_exp2f(x * TWO_LOG2E);
                float th = (ex - 1.f) * __builtin_amdgcn_rcpf(ex + 1.f);
                partial  = fmaf(vvx[c], th, partial);
            }
        }
        #pragma unroll
        for (int off = 16; off > 0; off >>= 1)
            partial += __shfl_xor(partial, off, 32);
        if (lane == 0)
            sc[s] = (mask[b * S + s] != 0) ? -INFINITY : partial;
    }
    __syncthreads();

    // block max
    float lm = -INFINITY;
    for (int s = tid; s < S; s += 256) lm = fmaxf(lm, sc[s]);
    red[tid] = lm; __syncthreads();
    for (int off = 128; off > 0; off >>= 1) {
        if (tid < off) red[tid] = fmaxf(red[tid], red[tid + off]);
        __syncthreads();
    }
    const float mx = red[0];
    __syncthreads();

    // exp + block sum (exp2(-inf)=0 handles masked lanes)
    const float LOG2E = 1.4426950408889634f;
    float ls = 0.f;
    for (int s = tid; s < S; s += 256) {
        float e = __builtin_amdgcn_exp2f((sc[s] - mx) * LOG2E);
        sc[s] = e;
        ls += e;
    }
    red[tid] = ls; __syncthreads();
    for (int off = 128; off > 0; off >>= 1) {
        if (tid < off) red[tid] += red[tid + off];
        __syncthreads();
    }
    const float inv = __builtin_amdgcn_rcpf(red[0]);
    __syncthreads();

    float* op = out + (size_t)bt * S;
    for (int s = tid; s < S; s += 256) op[s] = sc[s] * inv;
}

// ---------------------------------------------------------------------------
extern "C" void kernel_launch(void* const* d_in, const int* in_sizes, int n_in,
                              void* d_out, int out_size, void* d_ws, size_t ws_size,
                              hipStream_t stream) {
    (void)in_sizes; (void)n_in; (void)out_size; (void)ws_size;
    constexpr int B = 4, T = 256, S = 512, D = 512;

    const float* dec  = (const float*)d_in[0];   // [B,T,D]
    const float* enc  = (const float*)d_in[1];   // [B,S,2D]
    const int*   mask = (const int*)  d_in[2];   // [B,S]
    const float* We   = (const float*)d_in[3];   // [D,2D]
    const float* Wd   = (const float*)d_in[4];   // [D,D]
    const float* v    = (const float*)d_in[5];   // [D]
    float* out = (float*)d_out;                  // [B,T,S]

    float* e1 = (float*)d_ws;                    // [B*S, D] = 4 MB
    float* e2 = e1 + (size_t)B * S * D;          // [B*T, D] = 2 MB

    // e1: M=2048, N=512, K=1024 -> (128*32)/8 = 512 blocks
    gemm_awt_f32_wmma<<<512, 256, 0, stream>>>(enc, We, e1, B * S, D, 2 * D);
    // e2: M=1024, N=512, K=512  -> (64*32)/8  = 256 blocks
    gemm_awt_f32_wmma<<<256, 256, 0, stream>>>(dec, Wd, e2, B * T, D, D);
    // fused tanh-score + masked softmax, one block per (b,t)
    attn_tanh_softmax<<<B * T, 256, 0, stream>>>(e1, e2, mask, v, out);
}